// StreamQ_86388972192149
// MI455X (gfx1250) — compile-verified
//
#include <hip/hip_runtime.h>

// ---------------------------------------------------------------------------
// Fused StreamQ forward for MI455X (gfx1250, wave32).
// Single workgroup of 256 threads (8 wave32s) on one WGP; all intermediates
// live in LDS. fc1 (128x1024 matvec) runs on V_WMMA_F32_16X16X4_F32 with a
// broadcast-B layout so every D column carries the matvec result.
// ---------------------------------------------------------------------------

typedef __attribute__((ext_vector_type(2))) float v2f;
typedef __attribute__((ext_vector_type(8))) float v8f;

#define LN_EPS   1e-5f
#define NEG_SLP  0.01f

__device__ __forceinline__ float wave_reduce_add(float v) {
  // wave32 tree reduction (gfx1250 is wave32-only)
  #pragma unroll
  for (int off = 16; off > 0; off >>= 1) v += __shfl_down(v, off, 32);
  return v;
}

__global__ __launch_bounds__(256, 1)
void streamq_fused_kernel(const float* __restrict__ x,      // (10,10,4) HWC
                          const float* __restrict__ cw,     // (16,4,3,3)
                          const float* __restrict__ cb,     // (16,)
                          const float* __restrict__ w1,     // (128,1024)
                          const float* __restrict__ b1,     // (128,)
                          const float* __restrict__ w2,     // (3,128)
                          const float* __restrict__ b2,     // (3,)
                          float* __restrict__ out) {        // (3,)
  __shared__ float s_x[400];     // input image
  __shared__ float s_w[576];     // conv weights
  __shared__ float s_h[1024];    // conv out -> normalized activations
  __shared__ float s_h2[128];    // fc1 out -> normalized
  __shared__ float s_red[16];    // cross-wave reduction scratch
  __shared__ float s_mean, s_rstd;

  const int tid  = threadIdx.x;
  const int lane = tid & 31;
  const int wv   = tid >> 5;     // 0..7

  // --- Prefetch w1 (512 KB) into cache early: emits global_prefetch_b8 ---
  #pragma unroll
  for (int i = 0; i < 8; ++i)
    __builtin_prefetch(w1 + (tid * 8 + i) * 64, 0, 3);

  // --- Stage 0: stage x and conv weights through LDS ---
  for (int i = tid; i < 400; i += 256) s_x[i] = x[i];
  for (int i = tid; i < 576; i += 256) s_w[i] = cw[i];
  __syncthreads();

  // --- Stage 1: 3x3 VALID conv, 4->16 ch, out (16,8,8) = 1024 elems ---
  // 4 outputs per thread
  float hv[4];
  #pragma unroll
  for (int r = 0; r < 4; ++r) {
    const int p   = tid + 256 * r;
    const int oc  = p >> 6;
    const int rem = p & 63;
    const int oy  = rem >> 3;
    const int ox  = rem & 7;
    float s = cb[oc];
    #pragma unroll
    for (int ic = 0; ic < 4; ++ic)
      #pragma unroll
      for (int ky = 0; ky < 3; ++ky)
        #pragma unroll
        for (int kx = 0; kx < 3; ++kx)
          s += s_x[((oy + ky) * 10 + (ox + kx)) * 4 + ic] *
               s_w[oc * 36 + ic * 9 + ky * 3 + kx];
    hv[r] = s;
    s_h[p] = s;
  }

  // --- Stage 2: layernorm over all 1024 + leaky relu ---
  {
    float ps = 0.f, pq = 0.f;
    #pragma unroll
    for (int r = 0; r < 4; ++r) { ps += hv[r]; pq += hv[r] * hv[r]; }
    ps = wave_reduce_add(ps);
    pq = wave_reduce_add(pq);
    if (lane == 0) { s_red[wv] = ps; s_red[8 + wv] = pq; }
    __syncthreads();
    if (tid == 0) {
      float S = 0.f, Q = 0.f;
      #pragma unroll
      for (int i = 0; i < 8; ++i) { S += s_red[i]; Q += s_red[8 + i]; }
      const float m = S * (1.0f / 1024.0f);
      const float v = Q * (1.0f / 1024.0f) - m * m;
      s_mean = m;
      s_rstd = rsqrtf(v + LN_EPS);
    }
    __syncthreads();
    const float m = s_mean, rs = s_rstd;
    #pragma unroll
    for (int r = 0; r < 4; ++r) {
      float t = (hv[r] - m) * rs;
      t = (t >= 0.f) ? t : NEG_SLP * t;
      s_h[tid + 256 * r] = t;
    }
  }
  __syncthreads();

  // --- Stage 3: fc1 = w1 @ h via V_WMMA_F32_16X16X4_F32 ---
  // Wave wv owns rows [16*wv, 16*wv+16). A per ISA 16x4 layout:
  //   lane L holds row M=L&15, K-pair {0,1} (L<16) or {2,3} (L>=16).
  // B: same K-pair of the activation vector broadcast to all 16 columns,
  // so every D column equals the matvec slice (layout-robust).
  {
    const int M   = lane & 15;
    const int hi  = lane >> 4;          // 0 or 1
    const int klo = hi << 1;            // 0 or 2
    const float* arow = w1 + (wv * 16 + M) * 1024 + klo;

    v8f acc = {0.f, 0.f, 0.f, 0.f, 0.f, 0.f, 0.f, 0.f};
    #pragma unroll 4
    for (int k0 = 0; k0 < 1024; k0 += 4) {
      v2f a = *(const v2f*)(arow + k0);       // global_load_b64, 8B aligned
      v2f b;
      b.x = s_h[k0 + klo];
      b.y = s_h[k0 + klo + 1];
      acc = __builtin_amdgcn_wmma_f32_16x16x4_f32(
          /*neg_a=*/false, a, /*neg_b=*/false, b,
          /*c_mod=*/(short)0, acc, /*reuse_a=*/false, /*reuse_b=*/false);
    }

    // D layout: VGPR v holds row v (lanes 0-15) / row v+8 (lanes 16-31);
    // all 16 columns identical -> lanes 0 and 16 write rows 0-7 / 8-15.
    if ((lane & 15) == 0) {
      #pragma unroll
      for (int v = 0; v < 8; ++v) {
        const int r = wv * 16 + hi * 8 + v;
        s_h2[r] = acc[v] + b1[r];
      }
    }
  }
  __syncthreads();

  // --- Stage 4: layernorm over 128 + leaky relu ---
  {
    float t = (tid < 128) ? s_h2[tid] : 0.f;
    float ps = wave_reduce_add(t);
    float pq = wave_reduce_add(t * t);
    if (lane == 0) { s_red[wv] = ps; s_red[8 + wv] = pq; }
    __syncthreads();
    if (tid == 0) {
      float S = 0.f, Q = 0.f;
      #pragma unroll
      for (int i = 0; i < 8; ++i) { S += s_red[i]; Q += s_red[8 + i]; }
      const float m = S * (1.0f / 128.0f);
      const float v = Q * (1.0f / 128.0f) - m * m;
      s_mean = m;
      s_rstd = rsqrtf(v + LN_EPS);
    }
    __syncthreads();
    if (tid < 128) {
      float u = (s_h2[tid] - s_mean) * s_rstd;
      u = (u >= 0.f) ? u : NEG_SLP * u;
      s_h2[tid] = u;
    }
  }
  __syncthreads();

  // --- Stage 5: fc2 = w2 @ h2 + b2, 3 outputs, one wave each ---
  if (wv < 3) {
    float s = 0.f;
    for (int k = lane; k < 128; k += 32) s += w2[wv * 128 + k] * s_h2[k];
    s = wave_reduce_add(s);
    if (lane == 0) out[wv] = s + b2[wv];
  }
}

extern "C" void kernel_launch(void* const* d_in, const int* in_sizes, int n_in,
                              void* d_out, int out_size, void* d_ws, size_t ws_size,
                              hipStream_t stream) {
  (void)in_sizes; (void)n_in; (void)d_ws; (void)ws_size; (void)out_size;
  const float* x   = (const float*)d_in[0];
  const float* cw  = (const float*)d_in[1];
  const float* cb  = (const float*)d_in[2];
  const float* w1  = (const float*)d_in[3];
  const float* b1  = (const float*)d_in[4];
  const float* w2  = (const float*)d_in[5];
  const float* b2  = (const float*)d_in[6];
  float* out = (float*)d_out;

  streamq_fused_kernel<<<1, 256, 0, stream>>>(x, cw, cb, w1, b1, w2, b2, out);
}